// GraphSAGEConv_40716289966349
// MI455X (gfx1250) — compile-verified
//
#include <hip/hip_runtime.h>

// GraphSAGE conv for MI455X (gfx1250, wave32).
//   out = relu(concat(X, mean_nbr(X)) @ W + b) * drop_mask * 2.0
// Matrix op: V_WMMA_F32_16X16X4_F32 (exact fp32 — matches the fp32 reference,
// and the 3.3 GFLOP GEMM is far below both WMMA and HBM roofs, so no
// precision downgrade is warranted).

#define NN    50000      // nodes  (50000 = 3125 * 16, exact 16-row tiles)
#define EE    800000     // edges
#define DD    128        // input feature dim
#define DOUT  128        // output feature dim
#define KK    256        // concat dim = 2*DD
#define KEEP_SCALE 2.0f

typedef __attribute__((ext_vector_type(2))) float v2f;
typedef __attribute__((ext_vector_type(8))) float v8f;

// ---------------------------------------------------------------- zero scratch
__global__ __launch_bounds__(256) void sage_zero(float* __restrict__ p, int n) {
    int i = blockIdx.x * 256 + threadIdx.x;
    if (i < n) p[i] = 0.0f;
}

// ---------------------------------------------------------------- degree count
__global__ __launch_bounds__(256) void sage_deg(const int* __restrict__ dst,
                                                float* __restrict__ deg, int e) {
    int i = blockIdx.x * 256 + threadIdx.x;
    if (i < e) atomicAdd(deg + dst[i], 1.0f);
}

// ------------------------------------------------- edge scatter: nbr_sum += X[src]
// One wave32 handles 32 consecutive features of an edge -> coalesced gather
// from X (L2-resident, 25.6 MB << 192 MB L2) and coalesced f32 atomics into
// nbr_sum (also L2-resident).
__global__ __launch_bounds__(256) void sage_scatter(const float* __restrict__ X,
                                                    const int* __restrict__ src,
                                                    const int* __restrict__ dst,
                                                    float* __restrict__ nbr_sum) {
    long long i = (long long)blockIdx.x * 256 + threadIdx.x;   // i in [0, E*128)
    int e = (int)(i >> 7);
    int f = (int)(i & 127);
    if (e >= EE) return;
    int s = src[e];
    int d = dst[e];
    atomicAdd(nbr_sum + (long long)d * DD + f, X[(long long)s * DD + f]);
}

// ------------------------------------------------- fused WMMA GEMM epilogue
// Block: 256 threads (8 waves). Block owns 16 rows; wave w owns columns
// [16w, 16w+16). K = 256 swept in steps of 4 with v_wmma_f32_16x16x4_f32.
__global__ __launch_bounds__(256) void sage_gemm(const float* __restrict__ X,
                                                 const float* __restrict__ W,
                                                 const float* __restrict__ bias,
                                                 const int*   __restrict__ mask,
                                                 const float* __restrict__ nbr_sum,
                                                 const float* __restrict__ deg,
                                                 float* __restrict__ out) {
    // +4 column pad: A-operand LDS reads walk rows at stride 260 floats,
    // bank = (4*row + k) mod 64 -> 32 distinct banks across the wave.
    __shared__ float Ht[16][KK + 4];

    const int row0 = blockIdx.x * 16;
    const int t    = threadIdx.x;

    // Stage H tile: thread t fills column t of all 16 rows (coalesced across t).
    // Columns >= 128 are the mean-aggregated neighbor features (division fused).
    if (t < DD) {
        #pragma unroll
        for (int r = 0; r < 16; ++r)
            Ht[r][t] = X[(long long)(row0 + r) * DD + t];
    } else {
        const int c = t - DD;
        #pragma unroll
        for (int r = 0; r < 16; ++r) {
            float s  = nbr_sum[(long long)(row0 + r) * DD + c];
            float dg = deg[row0 + r];
            Ht[r][DD + c] = s / fmaxf(dg, 1.0f);
        }
    }
    __syncthreads();

    const int wave = t >> 5;
    const int lane = t & 31;
    const int half = lane >> 4;   // 0: lanes 0-15, 1: lanes 16-31
    const int l16  = lane & 15;
    const int col0 = wave * 16;

    v8f acc = {};

    // ISA 16x4 f32 A layout: lanes 0-15 hold (M=l16, K=k..k+1) in v0/v1,
    // lanes 16-31 hold (M=l16, K=k+2..k+3). B (4x16) mirrors with N across lanes.
    #pragma unroll 4
    for (int k = 0; k < KK; k += 4) {
        const int ka = k + 2 * half;
        v2f a, b;
        a.x = Ht[l16][ka];
        a.y = Ht[l16][ka + 1];
        b.x = W[(long long)ka       * DOUT + col0 + l16];
        b.y = W[(long long)(ka + 1) * DOUT + col0 + l16];
        acc = __builtin_amdgcn_wmma_f32_16x16x4_f32(
                  /*neg_a=*/false, a, /*neg_b=*/false, b,
                  /*c_mod=*/(short)0, acc, /*reuse_a=*/false, /*reuse_b=*/false);
    }

    // C/D layout: lane half selects M block {r} vs {r+8}; N = col0 + l16.
    const int col  = col0 + l16;
    const float bc = bias[col];
    #pragma unroll
    for (int r = 0; r < 8; ++r) {
        const int row = row0 + 8 * half + r;
        float v = acc[r] + bc;
        v = fmaxf(v, 0.0f);
        v *= (float)mask[(long long)row * DOUT + col] * KEEP_SCALE;
        out[(long long)row * DOUT + col] = v;
    }
}

// ---------------------------------------------------------------- launch
extern "C" void kernel_launch(void* const* d_in, const int* in_sizes, int n_in,
                              void* d_out, int out_size, void* d_ws, size_t ws_size,
                              hipStream_t stream) {
    (void)in_sizes; (void)n_in; (void)out_size; (void)ws_size;

    const float* X    = (const float*)d_in[0];   // [N, 128]
    const float* W    = (const float*)d_in[1];   // [256, 128]
    const float* bias = (const float*)d_in[2];   // [128]
    const int*   src  = (const int*)  d_in[3];   // [E]
    const int*   dst  = (const int*)  d_in[4];   // [E]
    const int*   mask = (const int*)  d_in[5];   // [N, 128]
    float*       out  = (float*)d_out;           // [N, 128]

    float* nbr_sum = (float*)d_ws;               // N*128 floats
    float* deg     = nbr_sum + (long long)NN * DD; // N floats

    // 1) zero nbr_sum + deg (workspace is poisoned)
    {
        int n = NN * DD + NN;
        sage_zero<<<(n + 255) / 256, 256, 0, stream>>>(nbr_sum, n);
    }
    // 2) degree histogram
    sage_deg<<<(EE + 255) / 256, 256, 0, stream>>>(dst, deg, EE);
    // 3) feature scatter-add (E * 128 lanes)
    {
        long long total = (long long)EE * DD;
        int blocks = (int)((total + 255) / 256);
        sage_scatter<<<blocks, 256, 0, stream>>>(X, src, dst, nbr_sum);
    }
    // 4) fused WMMA GEMM + bias + relu + dropout-mask
    sage_gemm<<<NN / 16, 256, 0, stream>>>(X, W, bias, mask, nbr_sum, deg, out);
}